// MultiChev_B_27462020891070
// MI455X (gfx1250) — compile-verified
//
#include <hip/hip_runtime.h>

typedef long long i64;
typedef __attribute__((ext_vector_type(2))) float v2f;
typedef __attribute__((ext_vector_type(8))) float v8f;

#define DIN   64
#define DOUT  300
#define WPAD  320     // weight N-dim padded to multiple of 16
#define KTOT  192     // concat K: [x | tx1 | p2]
#define NSUB  4       // 4 x 16 = 64 output columns per wave
#define NWTILE 64
#define NWAVES_N (WPAD / NWTILE)   // 5 waves cover the padded N dim
#define WROW2  (WPAD * 2)          // floats per K-pair row in interleaved Wc

// ---------------- helpers: graph normalization ----------------

__global__ void k_zero(float* __restrict__ p, size_t n) {
  size_t i = (size_t)blockIdx.x * blockDim.x + threadIdx.x;
  if (i < n) p[i] = 0.f;
}

__global__ void k_deg(const i64* __restrict__ row, const float* __restrict__ ew,
                      float* __restrict__ deg, int E) {
  int e = blockIdx.x * blockDim.x + threadIdx.x;
  if (e < E) atomicAdd(&deg[(int)row[e]], ew[e]);
}

__global__ void k_dis(float* __restrict__ deg, int N) {
  int i = blockIdx.x * blockDim.x + threadIdx.x;
  if (i < N) {
    float d = deg[i];
    deg[i] = (d > 0.f) ? rsqrtf(d) : 0.f;   // in place: deg -> D^-1/2
  }
}

__global__ void k_what(const i64* __restrict__ row, const i64* __restrict__ col,
                       const float* __restrict__ ew, const float* __restrict__ dis,
                       float* __restrict__ wh, int E) {
  int e = blockIdx.x * blockDim.x + threadIdx.x;
  if (e < E) wh[e] = -dis[(int)row[e]] * ew[e] * dis[(int)col[e]];
}

// ---------------- sparse propagation: dst[col] += w_hat * src[row] ----------------
// 2 features per thread: one b64 gather + two f32 atomics; 64-float segments stay
// coalesced. x/tx1/p2 (77 MB total) are L2-resident, so gathers/atomics land in L2.

__global__ void k_prop(const i64* __restrict__ row, const i64* __restrict__ col,
                       const float* __restrict__ wh, const float* __restrict__ src,
                       float* __restrict__ dst, int E) {
  long long idx = (long long)blockIdx.x * blockDim.x + threadIdx.x;
  if (idx >= (long long)E * (DIN / 2)) return;
  int e = (int)(idx >> 5);
  int d = (int)(idx & 31) * 2;
  int r = (int)row[e];
  int c = (int)col[e];
  float w = wh[e];
  v2f v = *(const v2f*)(src + (size_t)r * DIN + d);
  float* dp = dst + (size_t)c * DIN + d;
  atomicAdd(dp,     v.x * w);
  atomicAdd(dp + 1, v.y * w);
}

// ---------------- fold the 6 weight mats + 3 biases into one matrix ----------------
// Logical [192 x 320], stored K-pair interleaved: element (k, j) lives at
// Wc[((k>>1)*WPAD + j)*2 + (k&1)], so each lane's (K,K+1) B pair is one b64 load.
// rows 0..63   (x)  : [W10 | W20 | W30 - W32]
// rows 64..127 (tx1): [ 0  | W21 | W31      ]
// rows 128..191(p2) : [ 0  |  0  | 2*W32    ]   (tx2 = 2*p2 - x folded in)

__global__ void k_buildw(const float* __restrict__ W10, const float* __restrict__ W20,
                         const float* __restrict__ W21, const float* __restrict__ W30,
                         const float* __restrict__ W31, const float* __restrict__ W32,
                         const float* __restrict__ b1,  const float* __restrict__ b2,
                         const float* __restrict__ b3,
                         float* __restrict__ Wc, float* __restrict__ bc) {
  int i = blockIdx.x * blockDim.x + threadIdx.x;
  if (i >= KTOT * WPAD) return;
  int k = i / WPAD;
  int j = i % WPAD;
  float v = 0.f;
  if (j < 100) {
    if (k < 64) v = W10[k * 100 + j];
  } else if (j < 200) {
    int jj = j - 100;
    if (k < 64)       v = W20[k * 100 + jj];
    else if (k < 128) v = W21[(k - 64) * 100 + jj];
  } else if (j < 300) {
    int jj = j - 200;
    if (k < 64)       v = W30[k * 100 + jj] - W32[k * 100 + jj];
    else if (k < 128) v = W31[(k - 64) * 100 + jj];
    else              v = 2.f * W32[(k - 128) * 100 + jj];
  }
  Wc[((size_t)(k >> 1) * WPAD + j) * 2 + (k & 1)] = v;
  if (k == 0) {
    float b = 0.f;
    if (j < 100) b = b1[j];
    else if (j < 200) b = b2[j - 100];
    else if (j < 300) b = b3[j - 200];
    bc[j] = b;
  }
}

// ---------------- WMMA GEMM: out[N,300] = [x|tx1|p2] @ Wc + bc ----------------
// One wave per 16M x 64N tile: 4 f32 accumulators, A fragment loaded once per
// k-step (b64) and reused by 4 back-to-back v_wmma_f32_16x16x4_f32; B fragments
// are single b64 loads thanks to the pair-interleaved weight layout. K-loop is
// three straight-line 64-K phases (x, tx1, p2). Native fp32 matrix path = exact
// fp32 semantics; the kernel is bandwidth-bound so downcasting buys nothing.
// A-fragment layout (ISA 7.12.2, 32-bit A 16x4): lane[3:0]=M, lane[4] selects
// the K-pair, 2 VGPRs = K,K+1. B mirrored (KxN). C/D: n=lane[3:0], m=v+8*lane[4].

__device__ __forceinline__ void gemm_phase(const float* __restrict__ src,
                                           const float* __restrict__ wpp,
                                           int mrow, int lh, v8f acc[NSUB]) {
  const float* arow = src + (size_t)mrow * DIN + lh * 2;
  const float* brow = wpp + (size_t)lh * WROW2;   // lane's K-pair row
#pragma unroll 4
  for (int k = 0; k < 64; k += 4) {
    v2f a = *(const v2f*)(arow + k);
    const float* bp = brow + (size_t)(k >> 1) * WROW2;
#pragma unroll
    for (int s = 0; s < NSUB; ++s) {
      v2f b = *(const v2f*)(bp + s * 32);
      acc[s] = __builtin_amdgcn_wmma_f32_16x16x4_f32(
          false, a, false, b, (short)0, acc[s], false, false);
    }
  }
}

__global__ __launch_bounds__(256) void k_gemm(
    const float* __restrict__ x, const float* __restrict__ tx1,
    const float* __restrict__ p2, const float* __restrict__ Wc,
    const float* __restrict__ bc, float* __restrict__ out, int N) {
  int wave = (int)((blockIdx.x * blockDim.x + threadIdx.x) >> 5);
  int lane = threadIdx.x & 31;
  int mt = wave / NWAVES_N;
  int nt = wave % NWAVES_N;
  if (mt * 16 >= N) return;             // uniform per wave: EXEC stays all-ones
  int m0 = mt * 16;
  int n0 = nt * NWTILE;
  int lm = lane & 15;
  int lh = lane >> 4;

  int mrow = m0 + lm;
  if (mrow >= N) mrow = N - 1;          // clamp (N=100000 divides 16; safety only)

  v8f acc[NSUB];
#pragma unroll
  for (int s = 0; s < NSUB; ++s) acc[s] = (v8f){};

  // phase pair-bases: pairs 0..31 (x), 32..63 (tx1), 64..95 (p2)
  const float* wcp = Wc + (size_t)(n0 + lm) * 2;

  __builtin_prefetch(tx1 + (size_t)mrow * DIN, 0, 1);        // warm next phase's A
  gemm_phase(x,   wcp,                        mrow, lh, acc);
  __builtin_prefetch(p2 + (size_t)mrow * DIN, 0, 1);
  gemm_phase(tx1, wcp + (size_t)32 * WROW2,   mrow, lh, acc);
  gemm_phase(p2,  wcp + (size_t)64 * WROW2,   mrow, lh, acc);

#pragma unroll
  for (int s = 0; s < NSUB; ++s) {
    int n = n0 + s * 16 + lm;
    if (n < DOUT) {
      float bn = bc[n];
#pragma unroll
      for (int v = 0; v < 8; ++v) {
        int m = m0 + v + lh * 8;
        if (m < N) out[(size_t)m * DOUT + n] = acc[s][v] + bn;
      }
    }
  }
}

// ---------------- launch ----------------

extern "C" void kernel_launch(void* const* d_in, const int* in_sizes, int n_in,
                              void* d_out, int out_size, void* d_ws, size_t ws_size,
                              hipStream_t stream) {
  const float* x   = (const float*)d_in[0];
  const i64*   ei  = (const i64*)d_in[1];      // int64 [2, E]
  const float* ew  = (const float*)d_in[2];
  const float* W10 = (const float*)d_in[3];
  const float* b1  = (const float*)d_in[4];
  const float* W20 = (const float*)d_in[5];
  const float* W21 = (const float*)d_in[6];
  const float* b2  = (const float*)d_in[7];
  const float* W30 = (const float*)d_in[8];
  const float* W31 = (const float*)d_in[9];
  const float* W32 = (const float*)d_in[10];
  const float* b3  = (const float*)d_in[11];

  const int N = in_sizes[0] / DIN;
  const int E = in_sizes[1] / 2;
  const i64* row = ei;
  const i64* col = ei + E;

  // workspace layout (deg|tx1|p2 contiguous so one zero-fill covers all three)
  float* ws  = (float*)d_ws;
  float* deg = ws;                               // N      (becomes dis in place)
  float* tx1 = deg + N;                          // N*64
  float* p2  = tx1 + (size_t)N * DIN;            // N*64
  float* wh  = p2  + (size_t)N * DIN;            // E
  float* Wc  = wh  + E;                          // 96*320*2 (pair-interleaved)
  float* bc  = Wc  + (size_t)KTOT * WPAD;        // 320
  float* outp = (float*)d_out;

  size_t nz = (size_t)N * (1 + 2 * DIN);
  k_zero<<<(unsigned)((nz + 255) / 256), 256, 0, stream>>>(ws, nz);

  k_deg<<<(E + 255) / 256, 256, 0, stream>>>(row, ew, deg, E);
  k_dis<<<(N + 255) / 256, 256, 0, stream>>>(deg, N);
  k_what<<<(E + 255) / 256, 256, 0, stream>>>(row, col, ew, deg, wh, E);

  long long pn = (long long)E * (DIN / 2);       // 2 features per thread
  unsigned pblocks = (unsigned)((pn + 255) / 256);
  k_prop<<<pblocks, 256, 0, stream>>>(row, col, wh, x, tx1, E);   // tx1 = L_hat x
  k_prop<<<pblocks, 256, 0, stream>>>(row, col, wh, tx1, p2, E);  // p2  = L_hat tx1

  k_buildw<<<(KTOT * WPAD + 255) / 256, 256, 0, stream>>>(
      W10, W20, W21, W30, W31, W32, b1, b2, b3, Wc, bc);

  int mtiles = (N + 15) / 16;
  long long waves = (long long)mtiles * NWAVES_N;
  unsigned gblocks = (unsigned)((waves + 7) / 8);   // 8 waves per 256-thread block
  k_gemm<<<gblocks, 256, 0, stream>>>(x, tx1, p2, Wc, bc, outp, N);
}